// DyGCN_85864986182006
// MI455X (gfx1250) — compile-verified
//
#include <hip/hip_runtime.h>
#include <cstddef>
#include <cstdint>

typedef __attribute__((ext_vector_type(2))) float v2f;
typedef __attribute__((ext_vector_type(4))) float v4f;
typedef __attribute__((ext_vector_type(8))) float v8f;

namespace {
constexpr int kB = 16;                 // batch
constexpr int kN = 512;                // nodes
constexpr int kF = 64;                 // F_in == F_out
constexpr int kE = 16;                 // embedding dim
constexpr int kCheb = 3;               // Chebyshev order K
constexpr int kKtot = kE * kCheb * kF; // 3072: contraction depth of fused conv
}

// D = A(16x4 f32) x B(4x16 f32) + C(16x16 f32), wave32, full fp32 matrix pipe.
__device__ __forceinline__ v8f wmma_f32(v2f a, v2f b, v8f c) {
  return __builtin_amdgcn_wmma_f32_16x16x4_f32(false, a, false, b, (short)0, c,
                                               false, false);
}

// Async global->LDS 16B copy (VGPR-bypassing DMA, tracked by ASYNCcnt).
// vdst = per-lane LDS byte offset (low 32 bits of the shared-aperture address),
// vaddr = 64-bit global address.
__device__ __forceinline__ void async_copy16(const float* lds_dst,
                                             const float* gsrc) {
  uint32_t lds_off = (uint32_t)(uintptr_t)lds_dst;
  asm volatile("global_load_async_to_lds_b128 %0, %1, off"
               :
               : "v"(lds_off), "v"(gsrc)
               : "memory");
}
__device__ __forceinline__ void wait_async0() {
  asm volatile("s_wait_asynccnt 0x0" ::: "memory");
}

// ---------------------------------------------------------------------------
// Stage 1: dinv[b,n] = (max(sum_m relu(A[b,n,m]) + 1, 1e-6))^-1/2
// ---------------------------------------------------------------------------
__global__ __launch_bounds__(256) void dinv_kernel(const float* __restrict__ A,
                                                   float* __restrict__ dinv) {
  const int row = blockIdx.x;                       // b*N + n
  const float* a = A + (size_t)row * kN;
  float s = 0.f;
  for (int i = threadIdx.x; i < kN; i += 256) {
    float v = a[i];
    s += (v > 0.f) ? v : 0.f;
  }
  __shared__ float red[256];
  red[threadIdx.x] = s;
  __syncthreads();
  for (int off = 128; off > 0; off >>= 1) {
    if (threadIdx.x < off) red[threadIdx.x] += red[threadIdx.x + off];
    __syncthreads();
  }
  if (threadIdx.x == 0) {
    float d = red[0] + 1.0f;                        // +1 from the identity diag
    d = fmaxf(d, 1e-6f);
    dinv[row] = 1.0f / sqrtf(d);
  }
}

// ---------------------------------------------------------------------------
// Stage 2: Ahat[b,n,m] = dinv[b,n] * (relu(A[b,n,m]) + (n==m)) * dinv[b,m]
// ---------------------------------------------------------------------------
__global__ __launch_bounds__(256) void ahat_kernel(const float* __restrict__ A,
                                                   const float* __restrict__ dinv,
                                                   float* __restrict__ Ahat) {
  const long long total4 = (long long)kB * kN * kN / 4;
  for (long long i4 = (long long)blockIdx.x * blockDim.x + threadIdx.x;
       i4 < total4; i4 += (long long)gridDim.x * blockDim.x) {
    const long long base = i4 * 4;
    const long long bn = base / kN;       // b*N + n
    const int m0 = (int)(base % kN);
    const int n = (int)(bn % kN);
    const long long b = bn / kN;
    const float dn = dinv[bn];
    v4f va = *(const v4f*)&A[base];
    v4f r;
#pragma unroll
    for (int j = 0; j < 4; ++j) {
      float v = va[j];
      v = (v > 0.f) ? v : 0.f;
      if (n == m0 + j) v += 1.0f;
      r[j] = dn * v * dinv[b * kN + m0 + j];
    }
    *(v4f*)&Ahat[base] = r;
  }
}

// ---------------------------------------------------------------------------
// Stage 3/4: batched f32 WMMA GEMM  C = A[MxK] @ B[KxN]
//   cheb_mode==1: C = 2*A@B - I  (Chebyshev T2)
// block = 256 threads = 8 waves; block tile 128(M) x 64(N); K chunk 16.
// Wave w owns a 32x32 subtile = 2x2 fragments. LDS tiles are K-minor
// (As[m][k], Bs[n][k], pitch 20): each WMMA fragment {k,k+1} is one aligned
// ds_load_b64 into an even VGPR pair, conflict-free across the 16 lanes.
// Double-buffered software pipeline: the async global->LDS DMA for tile i+1
// is issued before the WMMAs on tile i and only waited (s_wait_asynccnt 0)
// afterwards, so the copy runs in the shadow of the matrix work.
// All launch dims divide evenly (M%128==0, N%64==0, K%16==0).
// ---------------------------------------------------------------------------
__global__ __launch_bounds__(256) void gemm_wmma_kernel(
    const float* __restrict__ Ag, const float* __restrict__ Bg,
    float* __restrict__ Cg, int M, int Ncols, int K, int lda, int ldb, int ldc,
    long long strideA, long long strideB, long long strideC, int cheb_mode) {
  __shared__ alignas(16) float As[2][128][20];  // [buf][m][k], padded 16->20
  __shared__ alignas(16) float Bs[2][64][20];   // [buf][n][k], padded 16->20

  const float* A = Ag + strideA * blockIdx.z;
  const float* B = Bg + strideB * blockIdx.z;
  float* C = Cg + strideC * blockIdx.z;

  const int mBase0 = blockIdx.y * 128;
  const int nBase0 = blockIdx.x * 64;
  const int tid = threadIdx.x;
  const int w = tid >> 5;
  const int lane = tid & 31;
  const int r = lane & 15;
  const int h = lane >> 4;
  const int mSub = (w >> 1) * 32;            // 4 wave-rows of 32
  const int nSub = (w & 1) * 32;             // 2 wave-cols of 32

  v8f c00 = {}, c01 = {}, c10 = {}, c11 = {};

  // stage one K-chunk into LDS buffer `buf`
  auto stage = [&](int k0, int buf) {
    // A tile: 128x16 = 512 x 16B chunks, 2 per thread, async DMA to LDS
#pragma unroll
    for (int it = 0; it < 2; ++it) {
      const int i = tid + it * 256;
      const int mLoc = i >> 2;
      const int kLoc = (i & 3) * 4;
      async_copy16(&As[buf][mLoc][kLoc],
                   &A[(size_t)(mBase0 + mLoc) * lda + k0 + kLoc]);
    }
    // B tile: 16x64 (coalesced along n), transpose-scattered into [n][k]
    const int kLoc = tid >> 4;
    const int nLoc = (tid & 15) * 4;
    v4f vb = *(const v4f*)&B[(size_t)(k0 + kLoc) * ldb + nBase0 + nLoc];
    Bs[buf][nLoc + 0][kLoc] = vb.x;
    Bs[buf][nLoc + 1][kLoc] = vb.y;
    Bs[buf][nLoc + 2][kLoc] = vb.z;
    Bs[buf][nLoc + 3][kLoc] = vb.w;
  };

  stage(0, 0);               // prologue: fill buffer 0
  wait_async0();
  __syncthreads();

  const int nChunks = K >> 4;
  for (int ci = 0; ci < nChunks; ++ci) {
    if (ci + 1 < nChunks) stage((ci + 1) << 4, (ci + 1) & 1);  // prefetch i+1
    const int buf = ci & 1;

#pragma unroll
    for (int kk = 0; kk < 16; kk += 4) {
      const int kl = kk + 2 * h;             // ISA 16x16x4 frag: k = 2h+j
      v2f a0 = *(const v2f*)&As[buf][mSub + r][kl];
      v2f a1 = *(const v2f*)&As[buf][mSub + 16 + r][kl];
      v2f b0 = *(const v2f*)&Bs[buf][nSub + r][kl];
      v2f b1 = *(const v2f*)&Bs[buf][nSub + 16 + r][kl];
      c00 = wmma_f32(a0, b0, c00);
      c01 = wmma_f32(a0, b1, c01);
      c10 = wmma_f32(a1, b0, c10);
      c11 = wmma_f32(a1, b1, c11);
    }

    wait_async0();           // tile i+1 DMA done (overlapped with WMMAs above)
    __syncthreads();         // + B stores visible, buffer i fully consumed
  }

  // epilogue: 2x2 fragment stores; C/D frag layout: row = v + 8h, col = r
#pragma unroll
  for (int ti = 0; ti < 2; ++ti) {
#pragma unroll
    for (int tj = 0; tj < 2; ++tj) {
      const v8f& c = (ti == 0) ? ((tj == 0) ? c00 : c01)
                               : ((tj == 0) ? c10 : c11);
      const int col = nBase0 + nSub + tj * 16 + r;
#pragma unroll
      for (int v = 0; v < 8; ++v) {
        const int row = mBase0 + mSub + ti * 16 + v + 8 * h;
        float val = c[v];
        if (cheb_mode) val = 2.0f * val - ((row == col) ? 1.0f : 0.0f);
        C[(size_t)row * ldc + col] = val;
      }
    }
  }
}

// ---------------------------------------------------------------------------
// Stage 5: fused adaptive conv as a rank-3072 WMMA GEMM.
//   z[row, (e,k,f)] = emb[row,e] * xg_k[row,f]  (A-frag built on the fly)
//   out[row, o]     = z @ weights_pool_flat[3072,64] + emb @ bias_pool
// block = 256 threads = 8 waves staging 64 rows; wave w covers 32 rows
// (two 16-row A-frags sharing one B-frag -> 2 WMMA per pair of W loads)
// x 16 cols (col tile w&3). xg tiles staged once via async global->LDS DMA.
// ---------------------------------------------------------------------------
__global__ __launch_bounds__(256) void fused_out_kernel(
    const float* __restrict__ xg0, const float* __restrict__ xg1,
    const float* __restrict__ xg2, const float* __restrict__ emb,
    const float* __restrict__ W,   // weights_pool, flat [E*K*F, Fout]
    const float* __restrict__ bias, float* __restrict__ out) {
  __shared__ alignas(16) float emb_s[64][20];          // pad 20: conflict-free
  __shared__ alignas(16) float xg_s[3][64][68];        // pad 68: conflict-free

  const int node0 = blockIdx.x * 64;
  const int tid = threadIdx.x;

  // stage x_g tiles first (async DMA, 16B chunks; row pitch 272B keeps every
  // chunk 16B aligned), then emb with synchronous scalar stores — the DMA
  // runs in the shadow of the emb staging.
  const float* xgp[3] = {xg0, xg1, xg2};
#pragma unroll
  for (int k = 0; k < 3; ++k) {
    const float* src = xgp[k] + (size_t)node0 * kF;
    for (int i = tid; i < 64 * kF / 4; i += 256) {
      const int rl = i >> 4;                           // 16 float4 per row
      const int f4 = (i & 15) * 4;
      async_copy16(&xg_s[k][rl][f4], &src[rl * kF + f4]);
    }
  }
  for (int i = tid; i < 64 * kE; i += 256) {
    const int rl = i >> 4;
    const int e = i & 15;
    emb_s[rl][e] = emb[(size_t)(node0 + rl) * kE + e];
  }
  wait_async0();
  __syncthreads();

  const int w = tid >> 5;
  const int lane = tid & 31;
  const int r = lane & 15;
  const int h = lane >> 4;
  const int rowG = (w >> 2) * 32;                      // 2 groups of 32 rows
  const int col = (w & 3) * 16 + r;                    // 4 col tiles x 16

  v8f c0 = {}, c1 = {};
  // lane's contraction index: k1 = kk + 2h + j ; e/kc/f tracked incrementally
  int k1 = 2 * h;
  int e = 0, kc = 0, f = 2 * h;
  const float* embRow0 = &emb_s[rowG + r][0];
  const float* embRow1 = &emb_s[rowG + 16 + r][0];
#pragma unroll 4
  for (int kk = 0; kk < kKtot; kk += 4) {
    const float ev0 = embRow0[e];
    const float ev1 = embRow1[e];
    v2f xv0 = *(const v2f*)&xg_s[kc][rowG + r][f];       // f even -> aligned
    v2f xv1 = *(const v2f*)&xg_s[kc][rowG + 16 + r][f];
    v2f a0, a1, bf;
    a0[0] = ev0 * xv0[0];
    a0[1] = ev0 * xv0[1];
    a1[0] = ev1 * xv1[0];
    a1[1] = ev1 * xv1[1];
    bf[0] = W[(size_t)k1 * kF + col];
    bf[1] = W[(size_t)(k1 + 1) * kF + col];
    c0 = wmma_f32(a0, bf, c0);
    c1 = wmma_f32(a1, bf, c1);
    k1 += 4;
    f += 4;
    if (f >= kF) {
      f -= kF;
      if (++kc == kCheb) { kc = 0; ++e; }
    }
  }

  // bias epilogue: + emb @ bias_pool
  float bcol[kE];
#pragma unroll
  for (int q = 0; q < kE; ++q) bcol[q] = bias[q * kF + col];
#pragma unroll
  for (int ti = 0; ti < 2; ++ti) {
    const v8f& c = ti ? c1 : c0;
#pragma unroll
    for (int v = 0; v < 8; ++v) {
      const int rl = rowG + ti * 16 + v + 8 * h;
      float acc = c[v];
#pragma unroll
      for (int q = 0; q < kE; ++q) acc += emb_s[rl][q] * bcol[q];
      out[(size_t)(node0 + rl) * kF + col] = acc;
    }
  }
}

// ---------------------------------------------------------------------------
extern "C" void kernel_launch(void* const* d_in, const int* in_sizes, int n_in,
                              void* d_out, int out_size, void* d_ws,
                              size_t ws_size, hipStream_t stream) {
  const float* x   = (const float*)d_in[0];   // [B,N,F]
  const float* emb = (const float*)d_in[1];   // [B,N,E]
  const float* A   = (const float*)d_in[2];   // [B,N,N]
  const float* Wp  = (const float*)d_in[3];   // [E,K,F,F] == flat [3072,64]
  const float* bp  = (const float*)d_in[4];   // [E,F]
  float* out = (float*)d_out;

  // workspace partition (~38 MB)
  float* Ahat = (float*)d_ws;
  float* T2   = Ahat + (size_t)kB * kN * kN;
  float* dinv = T2 + (size_t)kB * kN * kN;
  float* xg1  = dinv + (size_t)kB * kN;
  float* xg2  = xg1 + (size_t)kB * kN * kF;

  dinv_kernel<<<kB * kN, 256, 0, stream>>>(A, dinv);
  ahat_kernel<<<2048, 256, 0, stream>>>(A, dinv, Ahat);

  // T2 = 2*Ahat@Ahat - I   (per-batch 512^3 f32 WMMA GEMM)
  gemm_wmma_kernel<<<dim3(kN / 64, kN / 128, kB), 256, 0, stream>>>(
      Ahat, Ahat, T2, kN, kN, kN, kN, kN, kN, (long long)kN * kN,
      (long long)kN * kN, (long long)kN * kN, 1);

  // xg1 = Ahat @ x ; xg2 = T2 @ x
  gemm_wmma_kernel<<<dim3(kF / 64, kN / 128, kB), 256, 0, stream>>>(
      Ahat, x, xg1, kN, kF, kN, kN, kF, kF, (long long)kN * kN,
      (long long)kN * kF, (long long)kN * kF, 0);
  gemm_wmma_kernel<<<dim3(kF / 64, kN / 128, kB), 256, 0, stream>>>(
      T2, x, xg2, kN, kF, kN, kN, kF, kF, (long long)kN * kN,
      (long long)kN * kF, (long long)kN * kF, 0);

  // fused adaptive conv + bias
  fused_out_kernel<<<(kB * kN) / 64, 256, 0, stream>>>(x, xg1, xg2, emb, Wp, bp,
                                                       out);
}